// QuantConv_PW_57827439673971
// MI455X (gfx1250) — compile-verified
//
#include <hip/hip_runtime.h>

// out[b,o,h,w] = sum_i relu(x[b,i,h,w] * w[o,i])
//             = sum_i [ relu(x)*relu(w) + relu(-x)*relu(-w) ]   (exact identity)
// => one fused f32 GEMM pair on V_WMMA_F32_16X16X4_F32 (exact f32 math).
//
// W pos/neg split is PRECOMPUTED into LDS as interleaved quads
// {wp[k],wp[k+1],wn[k],wn[k+1]} so one ds_load_b128 delivers both A fragments.
// XOR-swizzled slots keep the 16-lane-strided b128 reads bank-conflict-free
// with zero padding (LDS = 64 rows * 1024 B = 64 KB exactly).
//
// Block = 128 threads (4 wave32), tile 64(M) x 64(N): 256 blocks total for
// better WGP coverage on a tiny, latency-bound problem; per-wave mix is
// unchanged (8 wmma : 4 ds_load_b128 : 4 valu-relu per K-step).

typedef float v2f __attribute__((ext_vector_type(2)));
typedef float v4f __attribute__((ext_vector_type(4)));
typedef float v8f __attribute__((ext_vector_type(8)));

#define IN_PLANES  128
#define OUT_PLANES 256
#define HW         1024   // 32*32
#define ROW_DW     256    // 64 k-pair slots * 4 dwords per row

__device__ __forceinline__ float relu_pos(float x) {   // relu(x), 1 VALU op
    float r;
    asm("v_max_num_f32_e64 %0, %1, 0" : "=v"(r) : "v"(x));
    return r;
}
__device__ __forceinline__ float relu_neg(float x) {   // relu(-x), 1 VALU op
    float r;
    asm("v_max_num_f32_e64 %0, -%1, 0" : "=v"(r) : "v"(x));
    return r;
}

__global__ __launch_bounds__(128) void quantconv_pw_wmma(
    const float* __restrict__ x,   // (4, 128, 32, 32)
    const float* __restrict__ w,   // (256, 128)
    float* __restrict__ out)       // (4, 256, 32, 32)
{
    __shared__ float Wpn[64 * ROW_DW];   // 65,536 B

    const int tid  = threadIdx.x;
    const int lane = tid & 31;
    const int wave = tid >> 5;      // 0..3 -> N strip
    const int ln   = lane & 15;     // M (A) / N (B,C,D) index within tile
    const int hk   = lane >> 4;     // which K-pair of the 4-wide K step

    const int mbase  = blockIdx.y * 64;      // o-block base
    const int nblk   = blockIdx.x;           // 0..63
    const int b      = nblk >> 4;            // batch
    const int hwbase = (nblk & 15) * 64;     // hw-block base

    // ---- Stage W block: build {wp,wp,wn,wn} quads, XOR-swizzled slots ----
    #pragma unroll
    for (int j = 0; j < 16; ++j) {
        int idx = j * 128 + tid;        // 0..2047 float4 slots of raw W
        int row = idx >> 5;             // 32 float4 per 128-wide row
        int c4  = idx & 31;             // covers k = 4*c4 .. 4*c4+3
        const float4 v = *(const float4*)(w + (mbase + row) * IN_PLANES + c4 * 4);
        const int X = row & 15;
        const int s0 = (2 * c4) ^ X;
        const int s1 = (2 * c4 + 1) ^ X;
        *(v4f*)(&Wpn[row * ROW_DW + 4 * s0]) =
            v4f{relu_pos(v.x), relu_pos(v.y), relu_neg(v.x), relu_neg(v.y)};
        *(v4f*)(&Wpn[row * ROW_DW + 4 * s1]) =
            v4f{relu_pos(v.z), relu_pos(v.w), relu_neg(v.z), relu_neg(v.w)};
    }
    __syncthreads();

    // ---- K loop: D = Wp*Xp + Wn*Xn, K = 128 in steps of 4 ----
    v8f acc[4] = {v8f{0.f,0.f,0.f,0.f,0.f,0.f,0.f,0.f},
                  v8f{0.f,0.f,0.f,0.f,0.f,0.f,0.f,0.f},
                  v8f{0.f,0.f,0.f,0.f,0.f,0.f,0.f,0.f},
                  v8f{0.f,0.f,0.f,0.f,0.f,0.f,0.f,0.f}};

    // Column pointer for this lane's N position; K rows are stride HW apart.
    const float* xb = x + (size_t)b * IN_PLANES * HW + hwbase + wave * 16 + ln;
    // Per-lane LDS row base for mt=0 (rows mt*16+ln; mt adds 16 KB immediates).
    const float* arow = &Wpn[ln * ROW_DW];

    #pragma unroll 8
    for (int k0 = 0; k0 < IN_PLANES; k0 += 4) {
        const int k = k0 + 2 * hk;
        // B fragment (4x16): lane l -> col l&15, K pair {k, k+1}
        const float b0 = xb[(size_t)k * HW];
        const float b1 = xb[(size_t)(k + 1) * HW];
        const v2f bp = { relu_pos(b0), relu_pos(b1) };
        const v2f bn = { relu_neg(b0), relu_neg(b1) };

        // Swizzled slot: same XOR for all 4 M-tiles (row&15 == ln).
        const int sp = ((k0 >> 1) + hk) ^ ln;
        const float* abase = arow + 4 * sp;

        #pragma unroll
        for (int mt = 0; mt < 4; ++mt) {
            const v4f q = *(const v4f*)(abase + mt * 16 * ROW_DW);  // b128
            const v2f ap = { q.x, q.y };   // relu(W) pair
            const v2f an = { q.z, q.w };   // relu(-W) pair
            acc[mt] = __builtin_amdgcn_wmma_f32_16x16x4_f32(
                false, ap, false, bp, (short)0, acc[mt], false, false);
            acc[mt] = __builtin_amdgcn_wmma_f32_16x16x4_f32(
                false, an, false, bn, (short)0, acc[mt], false, false);
        }
    }

    // ---- Store: C/D layout — VGPR r: row r + 8*(lane>>4), col lane&15 ----
    const int mo = hk << 3;   // 0 or 8
    float* outp = out + ((size_t)b * OUT_PLANES + mbase) * HW + hwbase + wave * 16 + ln;
    #pragma unroll
    for (int mt = 0; mt < 4; ++mt) {
        #pragma unroll
        for (int r = 0; r < 8; ++r) {
            outp[(size_t)(mt * 16 + r + mo) * HW] = acc[mt][r];
        }
    }
}

extern "C" void kernel_launch(void* const* d_in, const int* in_sizes, int n_in,
                              void* d_out, int out_size, void* d_ws, size_t ws_size,
                              hipStream_t stream) {
    const float* x = (const float*)d_in[0];   // 4*128*32*32 f32
    const float* w = (const float*)d_in[1];   // 256*128 f32 (o, i)
    float* out = (float*)d_out;               // 4*256*32*32 f32

    dim3 grid(64, 4);   // 64 N-blocks (4 batches x 16 hw-blocks), 4 M-blocks
    dim3 block(128);
    quantconv_pw_wmma<<<grid, block, 0, stream>>>(x, w, out);
}